// SimpleGIN_44744969290328
// MI455X (gfx1250) — compile-verified
//
#include <hip/hip_runtime.h>
#include <hip/hip_bf16.h>

typedef float v2f __attribute__((ext_vector_type(2)));
typedef float v8f __attribute__((ext_vector_type(8)));

#define D 128
#define CLS 10
#define NLAYERS 4
#define MT 5               // M tiles (of 16 rows) per block in the WMMA GEMM

// ---------------- generic zero ----------------
__global__ void zero_kernel(float* __restrict__ p, long long n) {
    long long i = (long long)blockIdx.x * blockDim.x + threadIdx.x;
    long long stride = (long long)gridDim.x * blockDim.x;
    for (; i < n; i += stride) p[i] = 0.0f;
}

// ---------------- degree / norm ----------------
__global__ void deg_kernel(const int* __restrict__ dst, float* __restrict__ deg, int E) {
    int e = blockIdx.x * blockDim.x + threadIdx.x;
    if (e < E) atomicAdd(&deg[dst[e]], 1.0f);
}

__global__ void norm_kernel(float* __restrict__ nrm, int N) {
    int n = blockIdx.x * blockDim.x + threadIdx.x;
    if (n < N) {
        float d = nrm[n];            // holds degree
        d = d < 1.0f ? 1.0f : d;
        nrm[n] = rsqrtf(d);
    }
}

// ---------------- WMMA f32 GEMM: C[MxD] = A[MxD] @ W[DxD] + bias ----------------
// One block = 80-row M tile (5 register-blocked 16-row WMMA tiles, B frag reused 5x).
// 8 waves; wave w computes columns [16w,16w+16). M must be a multiple of 80.
__global__ __launch_bounds__(256) void gemm128_wmma(const float* __restrict__ A,
                                                    const float* __restrict__ W,
                                                    const float* __restrict__ bias,
                                                    float* __restrict__ C, int M) {
    const int lane = threadIdx.x & 31;
    const int wave = threadIdx.x >> 5;
    const int m0   = blockIdx.x * (16 * MT);
    const int row  = lane & 15;      // M index within tile (A) / N col (B,C,D)
    const int hi   = lane >> 4;      // half-wave select
    const int ncol = (wave << 4) + row;

    const float* arow = A + (size_t)(m0 + row) * D;
    v8f acc[MT];
    #pragma unroll
    for (int t = 0; t < MT; ++t) acc[t] = (v8f){};

    for (int k0 = 0; k0 < D; k0 += 4) {
        const int ka = k0 + 2 * hi;               // K pair handled by this half-wave
        // B 4x16 f32 frag: VGPR0 = W[ka][ncol], VGPR1 = W[ka+1][ncol]
        v2f b;
        b.x = W[(size_t)ka * D + ncol];
        b.y = W[(size_t)(ka + 1) * D + ncol];
        #pragma unroll
        for (int t = 0; t < MT; ++t) {
            // A 16x4 f32 frag: VGPR0 = A[row][ka], VGPR1 = A[row][ka+1]
            v2f a = *(const v2f*)(arow + (size_t)t * 16 * D + ka);
            acc[t] = __builtin_amdgcn_wmma_f32_16x16x4_f32(false, a, false, b,
                                                           (short)0, acc[t], false, false);
        }
    }
    const float bv = bias[ncol];
    #pragma unroll
    for (int t = 0; t < MT; ++t) {
        float* cp = C + (size_t)(m0 + t * 16) * D + ncol;
        #pragma unroll
        for (int v = 0; v < 8; ++v) {
            int m = v + 8 * hi;                   // C/D layout: VGPR v -> row v (+8 for hi lanes)
            cp[(size_t)m * D] = acc[t][v] + bv;
        }
    }
}

// ---------------- edge scatter: ne[dst] += h[src] * norm[src] ----------------
// one wave per edge; each lane handles 4 consecutive features
__global__ void scatter_kernel(const float* __restrict__ h, const float* __restrict__ nrm,
                               float* __restrict__ ne,
                               const int* __restrict__ src, const int* __restrict__ dst, int E) {
    long long t = (long long)blockIdx.x * blockDim.x + threadIdx.x;
    int e = (int)(t >> 5);
    if (e >= E) return;
    int c = (int)(t & 31) << 2;
    const int s = src[e];
    const int d = dst[e];
    const float ns = nrm[s];
    float4 v = *(const float4*)(h + (size_t)s * D + c);
    float* p = ne + (size_t)d * D + c;
    atomicAdd(p + 0, v.x * ns);
    atomicAdd(p + 1, v.y * ns);
    atomicAdd(p + 2, v.z * ns);
    atomicAdd(p + 3, v.w * ns);
}

// ---------------- ne = h_in + (h_in*norm + ne) * norm  (pre-BN activation) ----------------
__global__ void combine_kernel(const float* __restrict__ h, float* __restrict__ ne,
                               const float* __restrict__ nrm, long long total4) {
    long long i = (long long)blockIdx.x * blockDim.x + threadIdx.x;
    long long stride = (long long)gridDim.x * blockDim.x;
    const float4* h4 = (const float4*)h;
    float4* e4 = (float4*)ne;
    for (; i < total4; i += stride) {
        float s = nrm[i >> 5];                    // 32 float4 per node (D=128)
        float4 a = h4[i], c = e4[i];
        c.x = a.x + (a.x * s + c.x) * s;
        c.y = a.y + (a.y * s + c.y) * s;
        c.z = a.z + (a.z * s + c.z) * s;
        c.w = a.w + (a.w * s + c.w) * s;
        e4[i] = c;
    }
}

// ---------------- BN stats: stats[j] = sum, stats[128+j] = sumsq ----------------
// stride is a multiple of 128 so each thread's feature index is fixed
__global__ void bn_stats_kernel(const float* __restrict__ x, float* __restrict__ stats,
                                long long total) {
    long long i = (long long)blockIdx.x * blockDim.x + threadIdx.x;
    long long stride = (long long)gridDim.x * blockDim.x;
    int j = (int)(i & (D - 1));
    float s = 0.0f, sq = 0.0f;
    for (; i < total; i += stride) {
        float v = x[i];
        s += v;
        sq += v * v;
    }
    atomicAdd(&stats[j], s);
    atomicAdd(&stats[D + j], sq);
}

// ---------------- BN apply: y = (x - mu) * rsqrt(var+eps) * gamma + beta ----------------
__global__ void bn_apply_kernel(const float* __restrict__ x, float* __restrict__ y,
                                const float* __restrict__ stats,
                                const float* __restrict__ gamma, const float* __restrict__ beta,
                                long long total, float invN) {
    long long i = (long long)blockIdx.x * blockDim.x + threadIdx.x;
    long long stride = (long long)gridDim.x * blockDim.x;
    for (; i < total; i += stride) {
        int j = (int)(i & (D - 1));
        float mu = stats[j] * invN;
        float var = stats[D + j] * invN - mu * mu;
        float inv = rsqrtf(var + 1e-5f);
        y[i] = (x[i] - mu) * inv * gamma[j] + beta[j];
    }
}

// ---------------- column sums into pooled[layer*128 + j] ----------------
__global__ void colsum_kernel(const float* __restrict__ x, float* __restrict__ out,
                              long long total) {
    long long i = (long long)blockIdx.x * blockDim.x + threadIdx.x;
    long long stride = (long long)gridDim.x * blockDim.x;
    int j = (int)(i & (D - 1));
    float s = 0.0f;
    for (; i < total; i += stride) s += x[i];
    atomicAdd(&out[j], s);
}

// ---------------- readout: score[c] = sum_l pooled[l]/N @ W_read[l] + b_read[l] ----------------
__global__ void readout_kernel(const float* __restrict__ pooled, const float* __restrict__ Wr,
                               const float* __restrict__ br, float* __restrict__ out,
                               float invN) {
    int c = threadIdx.x;
    if (c >= CLS) return;
    float s = 0.0f;
    for (int l = 0; l <= NLAYERS; ++l) {
        float acc = br[l * CLS + c];
        #pragma unroll 8
        for (int f = 0; f < D; ++f)
            acc += pooled[l * D + f] * invN * Wr[((size_t)l * D + f) * CLS + c];
        s += acc;
    }
    out[c] = s;
}

extern "C" void kernel_launch(void* const* d_in, const int* in_sizes, int n_in,
                              void* d_out, int out_size, void* d_ws, size_t ws_size,
                              hipStream_t stream) {
    const float* feat   = (const float*)d_in[0];
    const int*   src    = (const int*)  d_in[1];
    const int*   dst    = (const int*)  d_in[2];
    const float* W_emb  = (const float*)d_in[3];
    const float* b_emb  = (const float*)d_in[4];
    const float* gamma  = (const float*)d_in[5];
    const float* beta   = (const float*)d_in[6];
    const float* W_pred = (const float*)d_in[7];
    const float* b_pred = (const float*)d_in[8];
    const float* W_read = (const float*)d_in[9];
    const float* b_read = (const float*)d_in[10];

    const int N = in_sizes[0] / D;       // 100000
    const int E = in_sizes[1];           // 1600000
    const long long ND = (long long)N * D;
    const long long ND4 = ND / 4;
    const float invN = 1.0f / (float)N;

    float* ws = (float*)d_ws;
    float* h      = ws;                  // N*D
    float* hn     = ws + ND;             // N*D  (BN output feeding GEMM)
    float* ne     = ws + 2 * ND;         // N*D  (neighbor accumulator / pre-BN)
    float* nrm    = ws + 3 * ND;         // N (degree, then norm)
    float* stats  = nrm + N;             // 256
    float* pooled = stats + 2 * D;       // (L+1)*128

    const int TB = 256;
    const int gElem = 960;                               // grid for grid-stride elementwise
    const int gEdge = (int)(((long long)E * 32 + TB - 1) / TB);
    const int gGemm = N / (16 * MT);                     // 1250 blocks, exact

    // degree -> norm
    zero_kernel<<<256, TB, 0, stream>>>(nrm, N);
    deg_kernel<<<(E + TB - 1) / TB, TB, 0, stream>>>(dst, nrm, E);
    norm_kernel<<<(N + TB - 1) / TB, TB, 0, stream>>>(nrm, N);

    // h = feat @ W_emb + b_emb
    gemm128_wmma<<<gGemm, TB, 0, stream>>>(feat, W_emb, b_emb, h, N);

    // pooled init + layer-0 pooling
    zero_kernel<<<2, TB, 0, stream>>>(pooled, (NLAYERS + 1) * D);
    colsum_kernel<<<gElem, TB, 0, stream>>>(h, pooled, ND);

    for (int l = 0; l < NLAYERS; ++l) {
        zero_kernel<<<gElem, TB, 0, stream>>>(ne, ND);
        scatter_kernel<<<gEdge, TB, 0, stream>>>(h, nrm, ne, src, dst, E);
        combine_kernel<<<gElem, TB, 0, stream>>>(h, ne, nrm, ND4);
        zero_kernel<<<1, TB, 0, stream>>>(stats, 2 * D);
        bn_stats_kernel<<<480, TB, 0, stream>>>(ne, stats, ND);
        bn_apply_kernel<<<gElem, TB, 0, stream>>>(ne, hn, stats, gamma, beta, ND, invN);
        gemm128_wmma<<<gGemm, TB, 0, stream>>>(hn, W_pred + (size_t)l * D * D,
                                               b_pred + (size_t)l * D, h, N);
        colsum_kernel<<<gElem, TB, 0, stream>>>(h, pooled + (size_t)(l + 1) * D, ND);
    }

    readout_kernel<<<1, 32, 0, stream>>>(pooled, W_read, b_read, (float*)d_out, invN);
}